// SelfSupervisedMambaModel_1812476199773
// MI455X (gfx1250) — compile-verified
//
#include <hip/hip_runtime.h>
#include <hip/hip_bf16.h>
#include <cstdint>

#define D_MODEL 1024
#define D_STATE 16
#define D_CONV  4
#define D_INNER 2048
#define DT_RANK 64
#define BATCH   4
#define SEQ     2048
#define NROW    (BATCH * SEQ)   // 8192 rows (b*l)

typedef __attribute__((ext_vector_type(2))) float v2f;
typedef __attribute__((ext_vector_type(4))) float v4f;
typedef __attribute__((ext_vector_type(8))) float v8f;

// ---------------------------------------------------------------------------
// Generic GEMM:  C[M,N] = A[M,K] * W[N,K]^T      (fp32, V_WMMA_F32_16X16X4_F32)
// Block tile 64x128, K-step 16, 256 threads = 8 waves (2 along M x 4 along N).
// Wave tile 32x32 -> 4 accumulators, 16 WMMA per K-block per wave.
// Double-buffered LDS staging via GLOBAL_LOAD_ASYNC_TO_LDS_B128 (ASYNCcnt).
// ---------------------------------------------------------------------------
#define BM 64
#define BN 128
#define BK 16
#define TPAD 20   // LDS tile row stride in floats: 80B (16B-aligned, bank-conflict-free)

#define WMMA_F32(a, b, c) \
    __builtin_amdgcn_wmma_f32_16x16x4_f32(false, (a), false, (b), (short)0, (c), false, false)

__global__ __launch_bounds__(256)
void gemm_nt_wmma(const float* __restrict__ A, const float* __restrict__ W,
                  float* __restrict__ C, int M, int N, int K,
                  int lda, int ldw, int ldc)
{
    __shared__ float As[2][BM][TPAD];   // [buf][m][k]  row-major
    __shared__ float Ws[2][BN][TPAD];   // [buf][n][k]  row-major (W rows)

    const int tid  = threadIdx.x;
    const int lane = tid & 31;           // wave32
    const int wave = tid >> 5;
    const int wm   = (wave & 1) * 32;    // 2 waves along M
    const int wn   = (wave >> 1) * 32;   // 4 waves along N

    const int blk_n = blockIdx.x * BN;
    const int blk_m = blockIdx.y * BM;

    // staging assignment: 256 threads; 16B (4-float) chunk per async op
    const int arow  = tid >> 2;          // 0..63   (A rows)
    const int chunk = (tid & 3) * 4;     // k offset: 0,4,8,12
    const int wrow0 = tid >> 2;          // 0..63   (W rows, first half)
    const int wrow1 = wrow0 + 64;        // 64..127 (W rows, second half)
    int wn0 = blk_n + wrow0; if (wn0 >= N) wn0 = N - 1;   // clamp: keep EXEC full
    int wn1 = blk_n + wrow1; if (wn1 >= N) wn1 = N - 1;   // (dup rows never stored)

    const float* aptr  = A + (size_t)(blk_m + arow) * lda + chunk;
    const float* wptr0 = W + (size_t)wn0 * ldw + chunk;
    const float* wptr1 = W + (size_t)wn1 * ldw + chunk;

    const int frag_m  = lane & 15;          // M/N index within fragment
    const int frag_k0 = (lane >> 4) * 2;    // lanes 0-15 -> K0/K1, 16-31 -> K2/K3

    v8f c00 = {}, c01 = {}, c10 = {}, c11 = {};

    const int nk = K / BK;

    // async stage of K-block kb into LDS buffer `buf` (3 async ops per wave)
    auto stage = [&](int kb, int buf) {
        const int koff = kb * BK;
        uint32_t la = (uint32_t)(uintptr_t)&As[buf][arow][chunk];
        uint32_t l0 = (uint32_t)(uintptr_t)&Ws[buf][wrow0][chunk];
        uint32_t l1 = (uint32_t)(uintptr_t)&Ws[buf][wrow1][chunk];
        const float* ga = aptr  + koff;
        const float* g0 = wptr0 + koff;
        const float* g1 = wptr1 + koff;
        asm volatile("global_load_async_to_lds_b128 %0, %1, off"
                     :: "v"(la), "v"(ga) : "memory");
        asm volatile("global_load_async_to_lds_b128 %0, %1, off"
                     :: "v"(l0), "v"(g0) : "memory");
        asm volatile("global_load_async_to_lds_b128 %0, %1, off"
                     :: "v"(l1), "v"(g1) : "memory");
    };

    stage(0, 0);   // prologue

    for (int i = 0; i < nk; ++i) {
        const int buf = i & 1;
        if (i + 1 < nk) {
            stage(i + 1, buf ^ 1);
            // in-order completion: <=3 outstanding means stage(i)'s 3 ops landed
            asm volatile("s_wait_asynccnt 0x3" ::: "memory");
        } else {
            asm volatile("s_wait_asynccnt 0x0" ::: "memory");
        }
        __syncthreads();   // make all waves' async writes visible

        #pragma unroll
        for (int kk = 0; kk < BK; kk += 4) {
            const int k = kk + frag_k0;
            v2f a0, a1, b0, b1;
            a0.x = As[buf][wm + frag_m][k];
            a0.y = As[buf][wm + frag_m][k + 1];
            a1.x = As[buf][wm + 16 + frag_m][k];
            a1.y = As[buf][wm + 16 + frag_m][k + 1];
            b0.x = Ws[buf][wn + frag_m][k];
            b0.y = Ws[buf][wn + frag_m][k + 1];
            b1.x = Ws[buf][wn + 16 + frag_m][k];
            b1.y = Ws[buf][wn + 16 + frag_m][k + 1];
            c00 = WMMA_F32(a0, b0, c00);
            c01 = WMMA_F32(a0, b1, c01);
            c10 = WMMA_F32(a1, b0, c10);
            c11 = WMMA_F32(a1, b1, c11);
        }
        __syncthreads();   // reads done before this buffer is overwritten
    }

    // ---- write back: C/D layout: VGPR v -> row v + 8*(lane>>4), col lane&15 ----
    const int r0  = blk_m + wm + (lane >> 4) * 8;
    const int cn0 = blk_n + wn + frag_m;
    const int cn1 = cn0 + 16;
    if (cn0 < N) {
        #pragma unroll
        for (int v = 0; v < 8; ++v) {
            C[(size_t)(r0 + v) * ldc + cn0]      = c00[v];
            C[(size_t)(r0 + 16 + v) * ldc + cn0] = c10[v];
        }
    }
    if (cn1 < N) {
        #pragma unroll
        for (int v = 0; v < 8; ++v) {
            C[(size_t)(r0 + v) * ldc + cn1]      = c01[v];
            C[(size_t)(r0 + 16 + v) * ldc + cn1] = c11[v];
        }
    }
}

// ---------------------------------------------------------------------------
// Depthwise causal conv (D_CONV=4) + SiLU.  Reads xi half of xz, writes xc.
// ---------------------------------------------------------------------------
__global__ __launch_bounds__(256)
void conv_silu_kernel(const float* __restrict__ xz, const float* __restrict__ cw,
                      const float* __restrict__ cb, float* __restrict__ xc)
{
    const int idx = blockIdx.x * 256 + threadIdx.x;   // over NROW*D_INNER
    const int d  = idx & (D_INNER - 1);
    const int bl = idx >> 11;                          // D_INNER = 2048
    const int b  = bl >> 11;                           // SEQ = 2048
    const int l  = bl & (SEQ - 1);
    float acc = cb[d];
    #pragma unroll
    for (int j = 0; j < D_CONV; ++j) {
        const int ll = l + j - (D_CONV - 1);
        if (ll >= 0)
            acc += xz[(size_t)(b * SEQ + ll) * (2 * D_INNER) + d] * cw[d * D_CONV + j];
    }
    xc[idx] = acc / (1.0f + __expf(-acc));   // silu
}

// ---------------------------------------------------------------------------
// delta = softplus(dtproj + b_dt)   (in place on dtproj buffer)
// ---------------------------------------------------------------------------
__global__ __launch_bounds__(256)
void softplus_bias_kernel(float* __restrict__ dl, const float* __restrict__ b_dt)
{
    const int idx = blockIdx.x * 256 + threadIdx.x;
    const int d = idx & (D_INNER - 1);
    const float v = dl[idx] + b_dt[d];
    dl[idx] = (v > 20.0f) ? v : log1pf(__expf(v));
}

// ---------------------------------------------------------------------------
// Selective scan. One lane owns one (b,d) channel; h[16] in VGPRs.
// B/C vectors (shared across d) broadcast through LDS each timestep.
// Fuses the u*D skip term and silu(z) gating into the output.
// ---------------------------------------------------------------------------
__global__ __launch_bounds__(256)
void scan_kernel(const float* __restrict__ dl,  const float* __restrict__ xc,
                 const float* __restrict__ xdbl, const float* __restrict__ xz,
                 const float* __restrict__ A_log, const float* __restrict__ Dv,
                 float* __restrict__ yb)
{
    __shared__ float sBC[32];                 // [0:16)=B_t, [16:32)=C_t
    const int tid = threadIdx.x;
    const int d   = blockIdx.x * 256 + tid;
    const int b   = blockIdx.y;

    float h[D_STATE];
    float Ar[D_STATE];
    #pragma unroll
    for (int s = 0; s < D_STATE; ++s) {
        h[s]  = 0.0f;
        Ar[s] = -__expf(A_log[d * D_STATE + s]);   // A = -exp(A_log)
    }
    const float Dd = Dv[d];

    for (int l = 0; l < SEQ; ++l) {
        const size_t bl = (size_t)b * SEQ + l;
        if (tid < 32) sBC[tid] = xdbl[bl * 96 + DT_RANK + tid];
        __syncthreads();

        const float dlt = dl[bl * D_INNER + d];
        const float u   = xc[bl * D_INNER + d];
        const float du  = dlt * u;
        float y = 0.0f;
        #pragma unroll
        for (int s = 0; s < D_STATE; ++s) {
            h[s] = __expf(dlt * Ar[s]) * h[s] + du * sBC[s];
            y   += h[s] * sBC[16 + s];
        }
        const float z = xz[bl * (2 * D_INNER) + D_INNER + d];
        y = (y + u * Dd) * (z / (1.0f + __expf(-z)));   // (+u*D) * silu(z)
        yb[bl * D_INNER + d] = y;
        __syncthreads();
    }
}

// ---------------------------------------------------------------------------
// LayerNorm (in place on GEMM output) + residual add.
// One block per (b,l) row of 1024 elements.
// ---------------------------------------------------------------------------
__global__ __launch_bounds__(256)
void layernorm_residual_kernel(float* __restrict__ out, const float* __restrict__ x,
                               const float* __restrict__ g, const float* __restrict__ bb)
{
    __shared__ float red[256];
    const int row = blockIdx.x;
    const int tid = threadIdx.x;
    const size_t base = (size_t)row * D_MODEL;

    float v[4];
    float s = 0.0f, sq = 0.0f;
    #pragma unroll
    for (int i = 0; i < 4; ++i) {
        v[i] = out[base + tid + i * 256];
        s  += v[i];
        sq += v[i] * v[i];
    }
    red[tid] = s;  __syncthreads();
    for (int o = 128; o > 0; o >>= 1) { if (tid < o) red[tid] += red[tid + o]; __syncthreads(); }
    const float mean = red[0] * (1.0f / D_MODEL);
    __syncthreads();
    red[tid] = sq; __syncthreads();
    for (int o = 128; o > 0; o >>= 1) { if (tid < o) red[tid] += red[tid + o]; __syncthreads(); }
    const float var = red[0] * (1.0f / D_MODEL) - mean * mean;
    const float rs  = rsqrtf(var + 1e-5f);
    #pragma unroll
    for (int i = 0; i < 4; ++i) {
        const int c = tid + i * 256;
        out[base + c] = (v[i] - mean) * rs * g[c] + bb[c] + x[base + c];
    }
}

// ---------------------------------------------------------------------------
extern "C" void kernel_launch(void* const* d_in, const int* in_sizes, int n_in,
                              void* d_out, int out_size, void* d_ws, size_t ws_size,
                              hipStream_t stream)
{
    (void)in_sizes; (void)n_in; (void)out_size; (void)ws_size;

    const float* x      = (const float*)d_in[0];
    const float* W_in   = (const float*)d_in[1];
    const float* conv_w = (const float*)d_in[2];
    const float* conv_b = (const float*)d_in[3];
    const float* W_x    = (const float*)d_in[4];
    const float* W_dt   = (const float*)d_in[5];
    const float* b_dt   = (const float*)d_in[6];
    const float* A_log  = (const float*)d_in[7];
    const float* Dv     = (const float*)d_in[8];
    const float* W_out  = (const float*)d_in[9];
    const float* ln_g   = (const float*)d_in[10];
    const float* ln_b   = (const float*)d_in[11];
    float* out = (float*)d_out;

    // workspace layout (floats):
    float* ws   = (float*)d_ws;
    float* xz   = ws;                                    // NROW * 4096
    float* xc   = xz   + (size_t)NROW * (2 * D_INNER);   // NROW * 2048
    float* xdbl = xc   + (size_t)NROW * D_INNER;         // NROW * 96
    float* dl   = xdbl + (size_t)NROW * 96;              // NROW * 2048
    float* yb   = dl   + (size_t)NROW * D_INNER;         // NROW * 2048

    const dim3 blk(256);

    // 1) xz = x @ W_in^T            (M=8192, N=4096, K=1024)
    gemm_nt_wmma<<<dim3((2 * D_INNER) / BN, NROW / BM), blk, 0, stream>>>(
        x, W_in, xz, NROW, 2 * D_INNER, D_MODEL, D_MODEL, D_MODEL, 2 * D_INNER);

    // 2) depthwise causal conv + silu  -> xc
    conv_silu_kernel<<<(NROW * D_INNER) / 256, blk, 0, stream>>>(xz, conv_w, conv_b, xc);

    // 3) x_dbl = xc @ W_x^T         (M=8192, N=96, K=2048)
    gemm_nt_wmma<<<dim3((96 + BN - 1) / BN, NROW / BM), blk, 0, stream>>>(
        xc, W_x, xdbl, NROW, 96, D_INNER, D_INNER, D_INNER, 96);

    // 4) dtproj = dt @ W_dt^T       (M=8192, N=2048, K=64; A rows stride 96)
    gemm_nt_wmma<<<dim3(D_INNER / BN, NROW / BM), blk, 0, stream>>>(
        xdbl, W_dt, dl, NROW, D_INNER, DT_RANK, 96, DT_RANK, D_INNER);

    // 5) delta = softplus(dtproj + b_dt)
    softplus_bias_kernel<<<(NROW * D_INNER) / 256, blk, 0, stream>>>(dl, b_dt);

    // 6) selective scan + D-skip + silu(z) gating -> yb
    scan_kernel<<<dim3(D_INNER / 256, BATCH), blk, 0, stream>>>(
        dl, xc, xdbl, xz, A_log, Dv, yb);

    // 7) out = yb @ W_out^T         (M=8192, N=1024, K=2048) -> d_out (pre-LN)
    gemm_nt_wmma<<<dim3(D_MODEL / BN, NROW / BM), blk, 0, stream>>>(
        yb, W_out, out, NROW, D_MODEL, D_INNER, D_INNER, D_INNER, D_MODEL);

    // 8) LayerNorm (in place) + residual
    layernorm_residual_kernel<<<NROW, blk, 0, stream>>>(out, x, ln_g, ln_b);
}